// PeriodicBoundary_13176959664673
// MI455X (gfx1250) — compile-verified
//
#include <hip/hip_runtime.h>

// Problem constants (static per reference)
#define NPTS   1024
#define IMAGES 27
#define MTOT   (IMAGES * NPTS)   // 27648 candidate images
#define KMAX   145               // MAX_NEIGHBOURS
#define CUT2   36.0f             // CUTOFF^2

typedef float v2f __attribute__((ext_vector_type(2)));
typedef float v8f __attribute__((ext_vector_type(8)));

// gfx1250 async memory->LDS path (ASYNCcnt-tracked), guarded so absence of the
// builtins falls back to the load/ds_store staging loop.
#if defined(__AMDGCN__) && __has_builtin(__builtin_amdgcn_global_load_async_to_lds_b128) && __has_builtin(__builtin_amdgcn_s_wait_asynccnt)
#define USE_ASYNC_LDS 1
// Exact parameter types per hipcc diagnostic: b128 variant takes
// (int4-vec addrspace(1)*, int4-vec addrspace(3)*, imm offset, imm cpol).
typedef int v4i_raw __attribute__((vector_size(16)));
typedef __attribute__((address_space(1))) v4i_raw* async_gptr;
typedef __attribute__((address_space(3))) v4i_raw* async_lptr;
#else
#define USE_ASYNC_LDS 0
#endif

__device__ __forceinline__ unsigned ballot32(bool p) {
#if __has_builtin(__builtin_amdgcn_ballot_w32)
    return __builtin_amdgcn_ballot_w32(p);
#else
    return (unsigned)__ballot(p);
#endif
}

__global__ __launch_bounds__(32) void pb_init_max(int* out_max) {
    if (threadIdx.x == 0) *out_max = 0;
}

// One wave32 per 16-centre tile. Iterates 27648/16 = 1728 neighbour tiles,
// each tile = one v_wmma_f32_16x16x4_f32 producing 16x16 squared distances.
__global__ __launch_bounds__(32) void pb_neigh_kernel(
    const float* __restrict__ positions,  // [1024,3] f32
    const float* __restrict__ cell,       // [3,3] f32
    const int*   __restrict__ cell_list,  // [27,3] i32
    int* __restrict__ out_nb,             // [1024,145]
    int* __restrict__ out_cells,          // [1024,145,3]
    int* __restrict__ out_max)            // [1]
{
    __shared__ float sPos[NPTS * 3];      // 12 KB: whole position array in LDS
    __shared__ float sGrid[IMAGES][4];    // image translation vectors (cell_list @ cell)
    __shared__ int   sCL[IMAGES][3];      // raw cell_list (for cell_indices output)
    __shared__ float sC[16][4];           // this tile's centres: x,y,z,r2

    const int lane = threadIdx.x;   // 0..31 (wave32)
    const int col  = lane & 15;
    const int half = lane >> 4;     // 0: lanes 0-15, 1: lanes 16-31
    const int tileBase = blockIdx.x * 16;

    // --- Stage positions (768 x float4 = 12KB) into LDS ---
#if USE_ASYNC_LDS
    {
        const float4* p4 = (const float4*)positions;
        float4* s4 = (float4*)sPos;
        #pragma unroll
        for (int i = 0; i < (NPTS * 3) / 4 / 32; ++i) {   // 24 async b128 per lane-slot
            int idx = i * 32 + lane;
            __builtin_amdgcn_global_load_async_to_lds_b128(
                (async_gptr)(p4 + idx),
                (async_lptr)(s4 + idx),
                /*offset=*/0, /*cpol=*/0);
        }
        __builtin_amdgcn_s_wait_asynccnt(0);
    }
#else
    {
        const float4* p4 = (const float4*)positions;
        float4* s4 = (float4*)sPos;
        #pragma unroll 4
        for (int i = lane; i < (NPTS * 3) / 4; i += 32) s4[i] = p4[i];
    }
#endif
    // --- grid_points = cell_list @ cell, and keep cell_list ints ---
    if (lane < IMAGES) {
        int i0 = cell_list[lane * 3 + 0];
        int i1 = cell_list[lane * 3 + 1];
        int i2 = cell_list[lane * 3 + 2];
        sCL[lane][0] = i0; sCL[lane][1] = i1; sCL[lane][2] = i2;
        float f0 = (float)i0, f1 = (float)i1, f2 = (float)i2;
        #pragma unroll
        for (int c = 0; c < 3; ++c)
            sGrid[lane][c] = f0 * cell[0 * 3 + c] + f1 * cell[1 * 3 + c] + f2 * cell[2 * 3 + c];
        sGrid[lane][3] = 0.0f;
    }
    // --- centre tile coords + squared norms ---
    if (lane < 16) {
        int g = tileBase + lane;
        float x = positions[g * 3 + 0];
        float y = positions[g * 3 + 1];
        float z = positions[g * 3 + 2];
        sC[lane][0] = x; sC[lane][1] = y; sC[lane][2] = z;
        sC[lane][3] = x * x + y * y + z * z;
    }
    __syncthreads();

    // --- A matrix (16x4 f32, per ISA layout): row M = col, K pair per half.
    //     A = -2 * centre coords (K component 3 is zero-padded). ---
    v2f A;
    if (half == 0) { A.x = -2.0f * sC[col][0]; A.y = -2.0f * sC[col][1]; }
    else           { A.x = -2.0f * sC[col][2]; A.y = 0.0f; }

    int cnt = 0;  // per-row neighbour count (meaningful for lanes 0..15)

    for (int jb = 0; jb < MTOT; jb += 16) {
        // Candidate column n = jb + col (same for both halves)
        int n   = jb + col;
        int img = n >> 10;
        int p   = n & (NPTS - 1);
        float px = sPos[p * 3 + 0] + sGrid[img][0];
        float py = sPos[p * 3 + 1] + sGrid[img][1];
        float pz = sPos[p * 3 + 2] + sGrid[img][2];
        float r2n = px * px + py * py + pz * pz;

        // B matrix (4x16 f32): row K = vgpr + 2*half, col = lane&15
        v2f B;
        if (half == 0) { B.x = px; B.y = py; }   // K=0 (x), K=1 (y)
        else           { B.x = pz; B.y = 0.0f; } // K=2 (z), K=3 (pad)

        // C matrix: element(row m = v + 8*half, col) = r2_centre[m] + r2_neigh[col]
        v8f C;
        #pragma unroll
        for (int v = 0; v < 8; ++v) C[v] = sC[v + (half << 3)][3] + r2n;

        v8f D;
#if __has_builtin(__builtin_amdgcn_wmma_f32_16x16x4_f32)
        // D = (-2*Ci) . Nj + (r2_i + r2_j) = ||Nj - Ci||^2   (16x16 tile)
        D = __builtin_amdgcn_wmma_f32_16x16x4_f32(
            /*neg_a=*/false, A, /*neg_b=*/false, B,
            /*c_mod=*/(short)0, C, /*reuse_a=*/false, /*reuse_b=*/false);
#else
        // Exact VALU fallback with identical tile layout
        #pragma unroll
        for (int v = 0; v < 8; ++v) {
            int m = v + (half << 3);
            float dx = px - sC[m][0], dy = py - sC[m][1], dz = pz - sC[m][2];
            D[v] = dx * dx + dy * dy + dz * dz;
        }
#endif
        // 8 ballots -> 16 row masks (bits 0-15: row v; bits 16-31: row v+8)
        unsigned b0 = ballot32(D[0] < CUT2);
        unsigned b1 = ballot32(D[1] < CUT2);
        unsigned b2 = ballot32(D[2] < CUT2);
        unsigned b3 = ballot32(D[3] < CUT2);
        unsigned b4 = ballot32(D[4] < CUT2);
        unsigned b5 = ballot32(D[5] < CUT2);
        unsigned b6 = ballot32(D[6] < CUT2);
        unsigned b7 = ballot32(D[7] < CUT2);
        unsigned anyb = b0 | b1 | b2 | b3 | b4 | b5 | b6 | b7;  // wave-uniform

        if (anyb) {
            if (lane < 16) {
                // Lane r owns output row tileBase+r; pick its 16-bit column mask.
                unsigned t = (lane & 4) ? ((lane & 2) ? ((lane & 1) ? b7 : b6)
                                                      : ((lane & 1) ? b5 : b4))
                                        : ((lane & 2) ? ((lane & 1) ? b3 : b2)
                                                      : ((lane & 1) ? b1 : b0));
                unsigned mask = (t >> ((lane >> 3) << 4)) & 0xFFFFu;
                int row = tileBase + lane;
                while (mask) {
                    int b = __builtin_ctz(mask);
                    mask &= mask - 1;
                    int f = jb + b;               // ascending flat index = argwhere order
                    if (cnt < KMAX) {
                        out_nb[row * KMAX + cnt] = f & (NPTS - 1);
                        int im2 = f >> 10;
                        int* oc = out_cells + (size_t)(row * KMAX + cnt) * 3;
                        oc[0] = sCL[im2][0]; oc[1] = sCL[im2][1]; oc[2] = sCL[im2][2];
                    }
                    cnt++;  // uncapped: matches reference neigh_mask.sum() max
                }
            }
        }
    }

    // Tail fill: -1 sentinel; cell_indices take() clips -1 -> cells[0] = cell_list[0]
    if (lane < 16) {
        int row = tileBase + lane;
        int c0 = sCL[0][0], c1 = sCL[0][1], c2 = sCL[0][2];
        int start = cnt < KMAX ? cnt : KMAX;
        for (int s = start; s < KMAX; ++s) {
            out_nb[row * KMAX + s] = -1;
            int* oc = out_cells + (size_t)(row * KMAX + s) * 3;
            oc[0] = c0; oc[1] = c1; oc[2] = c2;
        }
        atomicMax(out_max, cnt);
    }
}

extern "C" void kernel_launch(void* const* d_in, const int* in_sizes, int n_in,
                              void* d_out, int out_size, void* d_ws, size_t ws_size,
                              hipStream_t stream) {
    (void)in_sizes; (void)n_in; (void)out_size; (void)d_ws; (void)ws_size;
    const float* positions = (const float*)d_in[0];  // [1024,3] f32
    const float* cell      = (const float*)d_in[1];  // [3,3]    f32
    const int*   cell_list = (const int*)d_in[2];    // [27,3]   i32

    int* out       = (int*)d_out;                    // int32 outputs, concatenated
    int* out_nb    = out;                            // [1024,145]
    int* out_cells = out + NPTS * KMAX;              // [1024,145,3]
    int* out_max   = out + NPTS * KMAX * 4;          // scalar

    pb_init_max<<<1, 32, 0, stream>>>(out_max);
    pb_neigh_kernel<<<NPTS / 16, 32, 0, stream>>>(positions, cell, cell_list,
                                                  out_nb, out_cells, out_max);
}